// SparseConvolutionDownsample_66383014527055
// MI455X (gfx1250) — compile-verified
//
#include <hip/hip_runtime.h>

typedef __attribute__((ext_vector_type(2))) float v2f;
typedef __attribute__((ext_vector_type(8))) float v8f;

#define CIN 32
#define COUT 64
#define NK 8
#define BN_EPS 1e-4f

// ---------------- K0: zero output accumulator + stats scratch ----------------
__global__ void k0_zero(float* __restrict__ out, float* __restrict__ ws, long long total) {
    long long i = (long long)blockIdx.x * blockDim.x + threadIdx.x;
    long long stride = (long long)gridDim.x * blockDim.x;
    for (; i < total; i += stride) out[i] = 0.0f;
    if (blockIdx.x == 0 && threadIdx.x < 129) ws[threadIdx.x] = 0.0f; // sum[64], sumsq[64], cnt
}

// ---------------- K1: masked WMMA GEMM + segment-sum scatter ----------------
// One wave handles a 16-point tile: A = 16x32 features (rows masked per tap k),
// B = 32x64 = W[k] staged transposed in LDS, D accumulated over 8 taps, then
// atomically scattered to out[seg[row]].
__global__ __launch_bounds__(256) void k1_conv(
    const float* __restrict__ feats, const float* __restrict__ W,
    const int* __restrict__ seg, const int* __restrict__ kidx,
    float* __restrict__ out, int N)
{
    __shared__ float ldsW[NK * COUT * CIN];   // 64 KB, layout [k][cout][cin]

    // cooperative transposed load of all 8 taps of W (coalesced global reads)
    for (int i = threadIdx.x; i < NK * CIN * COUT; i += 256) {
        int k    = i >> 11;        // /(32*64)
        int rem  = i & 2047;
        int cin  = rem >> 6;
        int cout = rem & 63;
        ldsW[(k * COUT + cout) * CIN + cin] = W[i];
    }
    __syncthreads();

    const int lane      = threadIdx.x & 31;
    const int wave      = threadIdx.x >> 5;
    const int rowInTile = lane & 15;          // M (A) / N column (B,D)
    const int half      = lane >> 4;          // K-half selector per ISA A layout
    const int base      = (blockIdx.x * 8 + wave) * 16;
    const int row       = base + rowInTile;

    // A fragments: 8 chunks of 16x4 fp32; lane holds K = kc*4 + half*2 + {0,1}
    v2f a[8];
    int kv = -1;
    if (row < N) {
        kv = kidx[row];
        #pragma unroll
        for (int kc = 0; kc < 8; ++kc)
            a[kc] = *reinterpret_cast<const v2f*>(
                feats + (long long)row * CIN + kc * 4 + half * 2);
    } else {
        #pragma unroll
        for (int kc = 0; kc < 8; ++kc) a[kc] = (v2f){0.0f, 0.0f};
    }

    v8f acc[4];
    #pragma unroll
    for (int nt = 0; nt < 4; ++nt) acc[nt] = (v8f){};

    const int bkk = half * 2;                 // B K-offset within a 4-chunk

    #pragma unroll 1                          // keep tap loop rolled (code size)
    for (int k = 0; k < NK; ++k) {
        const float m = (kv == k) ? 1.0f : 0.0f;   // row mask: y = sum_k mask_k(F) @ W[k]
        #pragma unroll
        for (int kc = 0; kc < 8; ++kc) {
            const v2f am = a[kc] * m;
            #pragma unroll
            for (int nt = 0; nt < 4; ++nt) {
                const v2f b = *reinterpret_cast<const v2f*>(
                    &ldsW[(k * COUT + nt * 16 + rowInTile) * CIN + kc * 4 + bkk]);
                acc[nt] = __builtin_amdgcn_wmma_f32_16x16x4_f32(
                    false, am, false, b, (short)0, acc[nt], false, false);
            }
        }
    }

    // D layout: VGPR r -> M = half*8 + r, N = nt*16 + (lane&15). Scatter-add per seg.
    int segv[8];
    #pragma unroll
    for (int r = 0; r < 8; ++r) {
        int orow = base + half * 8 + r;
        segv[r] = (orow < N) ? seg[orow] : -1;
    }
    #pragma unroll
    for (int nt = 0; nt < 4; ++nt) {
        const int col = nt * 16 + rowInTile;
        #pragma unroll
        for (int r = 0; r < 8; ++r) {
            if (segv[r] >= 0)
                atomicAdd(out + (long long)segv[r] * COUT + col, acc[nt][r]);
        }
    }
}

// ---------------- K2: per-column sum / sumsq / active count ----------------
// Inactive rows of out are exactly zero, so unweighted column sums equal the
// active-masked sums of the reference; only cnt needs the active[] mask.
__global__ __launch_bounds__(256) void k2_stats(
    const float* __restrict__ out, const float* __restrict__ active,
    float* __restrict__ ws, int N)
{
    const int col = threadIdx.x & 63;
    const int rg  = threadIdx.x >> 6;         // 4 row-groups per block
    float s = 0.f, ss = 0.f, cnt = 0.f;
    for (int row = blockIdx.x * 4 + rg; row < N; row += gridDim.x * 4) {
        float x = out[(long long)row * COUT + col];
        s += x; ss += x * x;
        if (col == 0) cnt += active[row];
    }
    __shared__ float red[256];
    red[threadIdx.x] = s;  __syncthreads();
    if (rg == 0) s = red[col] + red[64 + col] + red[128 + col] + red[192 + col];
    __syncthreads();
    red[threadIdx.x] = ss; __syncthreads();
    if (rg == 0) ss = red[col] + red[64 + col] + red[128 + col] + red[192 + col];
    __syncthreads();
    red[threadIdx.x] = cnt; __syncthreads();
    if (threadIdx.x == 0) {
        float c = red[0] + red[64] + red[128] + red[192];
        atomicAdd(ws + 128, c);
    }
    if (rg == 0) {
        atomicAdd(ws + col, s);
        atomicAdd(ws + 64 + col, ss);
    }
}

// ---------------- K3: finalize mean / fused scale ----------------
__global__ void k3_finalize(float* __restrict__ ws, const float* __restrict__ gamma) {
    int c = threadIdx.x;                      // 64 threads
    float cnt  = ws[128];
    float mean = ws[c] / cnt;
    float var  = fmaxf(ws[64 + c] / cnt - mean * mean, 0.0f);
    ws[192 + c] = mean;
    ws[256 + c] = __frsqrt_rn(var + BN_EPS) * gamma[c];
}

// ---------------- K4: normalize + ReLU + active mask (in place) ----------------
__global__ void k4_norm(float* __restrict__ out, const float* __restrict__ ws,
                        const float* __restrict__ beta, const float* __restrict__ active,
                        long long total)
{
    long long i = (long long)blockIdx.x * blockDim.x + threadIdx.x;
    long long stride = (long long)gridDim.x * blockDim.x;
    for (; i < total; i += stride) {
        int col = (int)(i & 63);
        long long row = i >> 6;
        float a = active[row];
        float v = (out[i] - ws[192 + col]) * ws[256 + col] + beta[col];
        out[i] = fmaxf(v, 0.0f) * a;
    }
}

extern "C" void kernel_launch(void* const* d_in, const int* in_sizes, int n_in,
                              void* d_out, int out_size, void* d_ws, size_t ws_size,
                              hipStream_t stream) {
    const float* feats  = (const float*)d_in[0];
    const float* W      = (const float*)d_in[1];
    const float* gamma  = (const float*)d_in[2];
    const float* beta   = (const float*)d_in[3];
    const float* active = (const float*)d_in[4];
    const int*   seg    = (const int*)d_in[5];
    const int*   kidx   = (const int*)d_in[6];

    const int N = in_sizes[5];                 // seg has one entry per point
    const long long total = (long long)N * COUT;
    float* out = (float*)d_out;
    float* ws  = (float*)d_ws;

    k0_zero<<<2048, 256, 0, stream>>>(out, ws, total);

    const int grid1 = (N + 127) / 128;         // 8 waves x 16 points per block
    k1_conv<<<grid1, 256, 0, stream>>>(feats, W, seg, kidx, out, N);

    k2_stats<<<1024, 256, 0, stream>>>(out, active, ws, N);
    k3_finalize<<<1, 64, 0, stream>>>(ws, gamma);
    k4_norm<<<4096, 256, 0, stream>>>(out, ws, beta, active, total);
}